// DecompGrid_6244882448586
// MI455X (gfx1250) — compile-verified
//
#include <hip/hip_runtime.h>
#include <math.h>
#include <stdint.h>

#define NFEAT 16

typedef unsigned int u32x4 __attribute__((ext_vector_type(4)));
typedef int          i32x8 __attribute__((ext_vector_type(8)));
typedef int          i32x4 __attribute__((ext_vector_type(4)));

__device__ __forceinline__ int iclamp(int v, int lo, int hi) {
    return v < lo ? lo : (v > hi ? hi : v);
}

// -------------------------------------------------------------------------
// Transpose one grid level from (C, R, R, R) to (R, R, R, C), C == 16.
// R is a compile-time power of two -> all index math is shifts/masks.
// Each 256-thread block handles a 16(c) x 16(x) tile for one (z, y) row.
// Source reads are one-time -> non-temporal (keep L2 for the HWC copies).
// -------------------------------------------------------------------------
template <int R>
__global__ __launch_bounds__(256) void transpose_chw_to_hwc(
    const float* __restrict__ src, float* __restrict__ dst)
{
    __shared__ float tile[16][17];  // +1 pad: avoid bank conflicts
    constexpr int NXC = R >> 4;     // number of 16-wide x chunks
    const int t  = blockIdx.x;
    const int xb = (t % NXC) << 4;
    const int y  = (t / NXC) % R;
    const int z  = t / (NXC * R);

    const int c  = threadIdx.x >> 4;
    const int xi = threadIdx.x & 15;
    tile[c][xi] = __builtin_nontemporal_load(
        &src[(size_t)(((c * R + z) * R + y) * R) + xb + xi]);
    __syncthreads();

    const int x2 = threadIdx.x >> 4;
    const int c2 = threadIdx.x & 15;
    dst[(size_t)(((z * R + y) * R + xb + x2)) * NFEAT + c2] = tile[c2][x2];
}

// -------------------------------------------------------------------------
// Trilinear gather for feature f. HWC==true: layout (z,y,x,16).
// HWC==false: reference layout (16,z,y,x). Matches torch grid_sample
// align_corners=True (floor before clip, weights from unclipped floor).
// -------------------------------------------------------------------------
template <bool HWC, typename GP>
__device__ __forceinline__ float trilerp16(GP g, int r, float cx, float cy,
                                           float cz, int f)
{
    const float s  = 0.5f * (float)(r - 1);
    const float fx = (cx + 1.0f) * s;
    const float fy = (cy + 1.0f) * s;
    const float fz = (cz + 1.0f) * s;
    const float xf = floorf(fx), yf = floorf(fy), zf = floorf(fz);
    const float wx = fx - xf, wy = fy - yf, wz = fz - zf;
    const int x0 = iclamp((int)xf, 0, r - 1);
    const int y0 = iclamp((int)yf, 0, r - 1);
    const int z0 = iclamp((int)zf, 0, r - 1);
    const int x1 = iclamp(x0 + 1, 0, r - 1);
    const int y1 = iclamp(y0 + 1, 0, r - 1);
    const int z1 = iclamp(z0 + 1, 0, r - 1);

    float v000, v001, v010, v011, v100, v101, v110, v111;
    if (HWC) {
        const int r00 = (z0 * r + y0) * r;
        const int r10 = (z0 * r + y1) * r;
        const int r01 = (z1 * r + y0) * r;
        const int r11 = (z1 * r + y1) * r;
        v000 = g[(size_t)(r00 + x0) * NFEAT + f];
        v001 = g[(size_t)(r00 + x1) * NFEAT + f];
        v010 = g[(size_t)(r10 + x0) * NFEAT + f];
        v011 = g[(size_t)(r10 + x1) * NFEAT + f];
        v100 = g[(size_t)(r01 + x0) * NFEAT + f];
        v101 = g[(size_t)(r01 + x1) * NFEAT + f];
        v110 = g[(size_t)(r11 + x0) * NFEAT + f];
        v111 = g[(size_t)(r11 + x1) * NFEAT + f];
    } else {
        const size_t cb = (size_t)f * r * r * r;
        const int r00 = (z0 * r + y0) * r;
        const int r10 = (z0 * r + y1) * r;
        const int r01 = (z1 * r + y0) * r;
        const int r11 = (z1 * r + y1) * r;
        v000 = g[cb + r00 + x0];
        v001 = g[cb + r00 + x1];
        v010 = g[cb + r10 + x0];
        v011 = g[cb + r10 + x1];
        v100 = g[cb + r01 + x0];
        v101 = g[cb + r01 + x1];
        v110 = g[cb + r11 + x0];
        v111 = g[cb + r11 + x1];
    }
    const float c00 = v000 * (1.0f - wx) + v001 * wx;
    const float c01 = v010 * (1.0f - wx) + v011 * wx;
    const float c10 = v100 * (1.0f - wx) + v101 * wx;
    const float c11 = v110 * (1.0f - wx) + v111 * wx;
    const float c0  = c00 * (1.0f - wy) + c01 * wy;
    const float c1  = c10 * (1.0f - wy) + c11 * wy;
    return c0 * (1.0f - wz) + c1 * wz;
}

// -------------------------------------------------------------------------
// Main kernel. 16 lanes per point (lane = feature). TRANSPOSED path:
// grids in (z,y,x,16) layout from workspace; level-0 grid (256 KB) is
// TDM-staged into LDS once per persistent block, then gathered via ds_load.
// Output is streaming (never re-read) -> non-temporal stores so the 512 MB
// of results do not evict the 153 MB L2-resident grid working set.
// -------------------------------------------------------------------------
template <bool TRANSPOSED>
__global__ __launch_bounds__(1024, 1) void decomp_grid_main(
    const float* __restrict__ xyz,
    const float* __restrict__ g0, const float* __restrict__ g1,
    const float* __restrict__ g2, const float* __restrict__ g3,
    float* __restrict__ out, int npts)
{
    extern __shared__ float lds0[];  // 256 KB: transposed level-0 grid

    if (TRANSPOSED) {
        if (threadIdx.x < 32) {  // wave 0 issues the TDM copy (EXEC ignored)
            const unsigned long long ga = (unsigned long long)(uintptr_t)g0;
            const unsigned ldsOff = (unsigned)__builtin_amdgcn_groupstaticsize();
            // D# group0: count=1 | lds_addr | global_addr[56:0] | type=2
            u32x4 d0 = { 1u, ldsOff, (unsigned)(ga & 0xFFFFFFFFu),
                         (unsigned)((ga >> 32) & 0x1FFFFFFu) | 0x80000000u };
            // D# group1: data_size=4B; tensor 4096x16, tile 4096x16, stride0=4096
            i32x8 d1 = { 0x00020000,          // data_size=2 (4B)
                         0x10000000,          // tensor_dim0 lo16=4096 in [31:16]
                         0x00100000,          // tensor_dim1 lo16=16  in [31:16]
                         0x10000000,          // tile_dim0=4096       in [31:16]
                         16,                  // tile_dim1=16, tile_dim2=0
                         4096,                // tensor_dim0_stride lo32
                         0, 0 };
            i32x4 dz4 = { 0, 0, 0, 0 };
            i32x8 dz8 = { 0, 0, 0, 0, 0, 0, 0, 0 };
            __builtin_amdgcn_tensor_load_to_lds(d0, d1, dz4, dz4, dz8, 0);
            __builtin_amdgcn_s_wait_tensorcnt(0);
        }
        __syncthreads();
    }

    const int f      = threadIdx.x & 15;                                  // feature lane
    const int gid    = (int)((blockIdx.x * blockDim.x + threadIdx.x) >> 4);
    const int stride = (int)((gridDim.x * blockDim.x) >> 4);

    for (int p = gid; p < npts; p += stride) {
        const float cx = xyz[(size_t)p * 3 + 0];
        const float cy = xyz[(size_t)p * 3 + 1];
        const float cz = xyz[(size_t)p * 3 + 2];
        float* o = out + (size_t)p * 64;
        float r0, r1, r2, r3;
        if (TRANSPOSED) {
            r0 = trilerp16<true>(lds0, 16, cx, cy, cz, f);  // LDS gather
            r1 = trilerp16<true>(g1, 32, cx, cy, cz, f);
            r2 = trilerp16<true>(g2, 64, cx, cy, cz, f);
            r3 = trilerp16<true>(g3, 128, cx, cy, cz, f);
        } else {
            r0 = trilerp16<false>(g0, 16, cx, cy, cz, f);
            r1 = trilerp16<false>(g1, 32, cx, cy, cz, f);
            r2 = trilerp16<false>(g2, 64, cx, cy, cz, f);
            r3 = trilerp16<false>(g3, 128, cx, cy, cz, f);
        }
        __builtin_nontemporal_store(r0, &o[f]);
        __builtin_nontemporal_store(r1, &o[16 + f]);
        __builtin_nontemporal_store(r2, &o[32 + f]);
        __builtin_nontemporal_store(r3, &o[48 + f]);
    }
}

extern "C" void kernel_launch(void* const* d_in, const int* in_sizes, int n_in,
                              void* d_out, int out_size, void* d_ws, size_t ws_size,
                              hipStream_t stream)
{
    const float* x  = (const float*)d_in[0];
    const float* g0 = (const float*)d_in[1];
    const float* g1 = (const float*)d_in[2];
    const float* g2 = (const float*)d_in[3];
    const float* g3 = (const float*)d_in[4];
    float* out = (float*)d_out;
    const int npts = in_sizes[0] / 3;

    const size_t e0 = (size_t)NFEAT * 16 * 16 * 16;
    const size_t e1 = (size_t)NFEAT * 32 * 32 * 32;
    const size_t e2 = (size_t)NFEAT * 64 * 64 * 64;
    const size_t e3 = (size_t)NFEAT * 128 * 128 * 128;
    const size_t need = (e0 + e1 + e2 + e3) * sizeof(float);

    if (ws_size >= need) {
        float* t0 = (float*)d_ws;
        float* t1 = t0 + e0;
        float* t2 = t1 + e1;
        float* t3 = t2 + e2;
        transpose_chw_to_hwc<16><<<16 * 16 * 1,     256, 0, stream>>>(g0, t0);
        transpose_chw_to_hwc<32><<<32 * 32 * 2,     256, 0, stream>>>(g1, t1);
        transpose_chw_to_hwc<64><<<64 * 64 * 4,     256, 0, stream>>>(g2, t2);
        transpose_chw_to_hwc<128><<<128 * 128 * 8,  256, 0, stream>>>(g3, t3);

        const int shmem = 256 * 1024;  // transposed g0 in LDS (320 KB max/WGP)
        (void)hipFuncSetAttribute((const void*)decomp_grid_main<true>,
                                  hipFuncAttributeMaxDynamicSharedMemorySize, shmem);
        decomp_grid_main<true><<<1024, 1024, shmem, stream>>>(
            x, t0, t1, t2, t3, out, npts);
    } else {
        // Fallback: gather directly from the reference (C,D,H,W) layout.
        decomp_grid_main<false><<<8192, 256, 0, stream>>>(
            x, g0, g1, g2, g3, out, npts);
    }
}